// MultiScaleGNN_35210141893311
// MI455X (gfx1250) — compile-verified
//
#include <hip/hip_runtime.h>

typedef float v2f __attribute__((ext_vector_type(2)));
typedef float v8f __attribute__((ext_vector_type(8)));

static constexpr int DIN = 128, DH = 128, DOUT = 64, S = 3, L = 3;

// ----------------------------------------------------------- hw fp32 atomic
// No-return GLOBAL_ATOMIC_ADD_F32 (tracked by STOREcnt); guarantees the
// single-instruction HW path instead of a possible CAS loop.
__device__ __forceinline__ void atomic_add_f32_hw(float* p, float v) {
  asm volatile("global_atomic_add_f32 %0, %1, off" ::"v"(p), "v"(v) : "memory");
}

// ---------------------------------------------------------------- utilities

__global__ void fill_kernel(float* __restrict__ p, long long n, float v) {
  long long i = blockIdx.x * (long long)blockDim.x + threadIdx.x;
  if (i < n) p[i] = v;
}

__global__ void deg_accum_kernel(const int* __restrict__ dst,
                                 float* __restrict__ deg, int E) {
  int e = blockIdx.x * blockDim.x + threadIdx.x;
  if (e < E) atomic_add_f32_hw(&deg[dst[e]], 1.0f);
}

__global__ void rsqrt_kernel(float* __restrict__ p, int n) {
  int i = blockIdx.x * blockDim.x + threadIdx.x;
  if (i < n) p[i] = rsqrtf(p[i]);
}

// ---------------------------------------------------- WMMA fp32 GEMM kernel
// C[M x NC] = A[M x K] @ W[K x NC] (+ bias).  W staged in LDS.
// Block = 256 threads = 8 waves; each wave owns a 16-row strip, NC/16 tiles.
template <int K, int NC>
__global__ __launch_bounds__(256) void gemm_wmma_kernel(
    const float* __restrict__ A, const float* __restrict__ W,
    const float* __restrict__ bias, float* __restrict__ C, int M) {
  constexpr int NT = NC / 16;
  __shared__ float sW[K * NC];
  for (int i = threadIdx.x; i < K * NC; i += 256) sW[i] = W[i];
  __syncthreads();

  const int lane = threadIdx.x & 31;
  const int wave = threadIdx.x >> 5;
  const int hl   = lane >> 4;   // 0: lanes 0-15, 1: lanes 16-31
  const int l16  = lane & 15;
  const int row0 = blockIdx.x * 128 + wave * 16;
  if (row0 >= M) return;        // M % 16 == 0, whole-wave guard

  v8f acc[NT] = {};
  const float* arow = A + (long long)(row0 + l16) * K;
  __builtin_prefetch(arow, 0, 3);            // global_prefetch_b8 (cacheline)
  __builtin_prefetch(arow + K / 2, 0, 3);

  for (int k = 0; k < K; k += 4) {
    // A 16x4 fp32 fragment: lanes 0-15 hold K=k..k+1, lanes 16-31 K=k+2..k+3
    v2f a;
    a.x = arow[k + hl * 2 + 0];
    a.y = arow[k + hl * 2 + 1];
#pragma unroll
    for (int t = 0; t < NT; ++t) {
      const int col = t * 16 + l16;
      v2f b;
      b.x = sW[(k + hl * 2 + 0) * NC + col];
      b.y = sW[(k + hl * 2 + 1) * NC + col];
      acc[t] = __builtin_amdgcn_wmma_f32_16x16x4_f32(
          /*neg_a=*/false, a, /*neg_b=*/false, b,
          /*c_mod=*/(short)0, acc[t], /*reuse_a=*/false, /*reuse_b=*/false);
    }
  }

#pragma unroll
  for (int t = 0; t < NT; ++t) {
    const int col = t * 16 + l16;
    const float badd = bias ? bias[col] : 0.0f;
#pragma unroll
    for (int r = 0; r < 8; ++r) {
      const int row = row0 + hl * 8 + r;      // C layout: VGPR r -> M=r / M=r+8
      C[(long long)row * NC + col] = acc[t][r] + badd;
    }
  }
}

// ------------------------------- agg init: agg = hw * dis^2 + b  (self-loop)
template <int D>
__global__ void init_selfloop_kernel(float* __restrict__ agg,
                                     const float* __restrict__ hw,
                                     const float* __restrict__ dis,
                                     const float* __restrict__ b, int Nn) {
  constexpr int CH = D / 4;
  long long t = blockIdx.x * (long long)blockDim.x + threadIdx.x;
  if (t >= (long long)Nn * CH) return;
  const int node = (int)(t / CH);
  const int c = (int)(t % CH);
  float dd = dis[node];
  dd *= dd;
  const float4 w = ((const float4*)hw)[t];
  const float4 bb = ((const float4*)b)[c];
  float4 o;
  o.x = fmaf(w.x, dd, bb.x);
  o.y = fmaf(w.y, dd, bb.y);
  o.z = fmaf(w.z, dd, bb.z);
  o.w = fmaf(w.w, dd, bb.w);
  ((float4*)agg)[t] = o;
}

// ---------------------------------------------- edge gather + atomic scatter
// agg[dst] += hw[src] * dis[src]*dis[dst]; one lane = one float4 chunk.
template <int D>
__global__ void edge_scatter_kernel(const float4* __restrict__ hw4,
                                    const int* __restrict__ src,
                                    const int* __restrict__ dst,
                                    const float* __restrict__ dis,
                                    float* __restrict__ agg, int E) {
  constexpr int CH = D / 4;
  long long t = blockIdx.x * (long long)blockDim.x + threadIdx.x;
  long long total = (long long)E * CH;
  if (t >= total) return;
  const int e = (int)(t / CH);
  const int c = (int)(t % CH);
  const int s = src[e];
  const int d = dst[e];
  const float coeff = dis[s] * dis[d];
  const float4 m = hw4[(long long)s * CH + c];
  float* base = agg + (long long)d * D + c * 4;
  atomic_add_f32_hw(base + 0, m.x * coeff);
  atomic_add_f32_hw(base + 1, m.y * coeff);
  atomic_add_f32_hw(base + 2, m.z * coeff);
  atomic_add_f32_hw(base + 3, m.w * coeff);
}

// ------------------------------- ReLU + LayerNorm in place (wave per node)
__global__ void relu_ln_kernel(float* __restrict__ h,
                               const float* __restrict__ g,
                               const float* __restrict__ be, int Nn) {
  const int gt = blockIdx.x * blockDim.x + threadIdx.x;
  const int node = gt >> 5;
  const int lane = gt & 31;
  if (node >= Nn) return;
  float4 v = ((const float4*)h)[node * 32 + lane];
  v.x = fmaxf(v.x, 0.0f);
  v.y = fmaxf(v.y, 0.0f);
  v.z = fmaxf(v.z, 0.0f);
  v.w = fmaxf(v.w, 0.0f);
  float sum = v.x + v.y + v.z + v.w;
  float sq = v.x * v.x + v.y * v.y + v.z * v.z + v.w * v.w;
#pragma unroll
  for (int off = 16; off > 0; off >>= 1) {
    sum += __shfl_xor(sum, off, 32);
    sq  += __shfl_xor(sq, off, 32);
  }
  const float mu  = sum * (1.0f / 128.0f);
  const float var = sq * (1.0f / 128.0f) - mu * mu;
  const float r = rsqrtf(var + 1e-5f);
  const float4 gg = ((const float4*)g)[lane];
  const float4 bt = ((const float4*)be)[lane];
  float4 o;
  o.x = (v.x - mu) * r * gg.x + bt.x;
  o.y = (v.y - mu) * r * gg.y + bt.y;
  o.z = (v.z - mu) * r * gg.z + bt.z;
  o.w = (v.w - mu) * r * gg.w + bt.w;
  ((float4*)h)[node * 32 + lane] = o;
}

// ------------------------- copy compact N x 64 block into fused (N x 192)
__global__ void copy_to_fused_kernel(const float* __restrict__ h,
                                     float* __restrict__ fused, int Nn,
                                     int colOff) {
  const int gt = blockIdx.x * blockDim.x + threadIdx.x;
  const int node = gt >> 4;  // 16 float4 chunks = 64 cols
  const int c = gt & 15;
  if (node >= Nn) return;
  const float4 v = ((const float4*)h)[node * 16 + c];
  *(float4*)(fused + (long long)node * (S * DOUT) + colOff + c * 4) = v;
}

// ------------------------------------------------------------------- driver
static inline int cdiv(long long a, long long b) { return (int)((a + b - 1) / b); }

extern "C" void kernel_launch(void* const* d_in, const int* in_sizes, int n_in,
                              void* d_out, int out_size, void* d_ws,
                              size_t ws_size, hipStream_t stream) {
  const float* x     = (const float*)d_in[0];
  const int*   ei    = (const int*)d_in[1];
  const float* W0    = (const float*)d_in[2];
  const float* b0    = (const float*)d_in[3];
  const float* W1    = (const float*)d_in[4];
  const float* b1    = (const float*)d_in[5];
  const float* W2    = (const float*)d_in[6];
  const float* b2    = (const float*)d_in[7];
  const float* gamma = (const float*)d_in[8];
  const float* beta  = (const float*)d_in[9];
  const float* Wf    = (const float*)d_in[10];
  const float* bf    = (const float*)d_in[11];
  float* out = (float*)d_out;

  const int N = in_sizes[0] / DIN;
  const int E = in_sizes[1] / 2;
  const int* srcI = ei;
  const int* dstI = ei + E;

  float* ws    = (float*)d_ws;
  float* dis   = ws;
  float* bufA  = dis + N;                       // N x 128 (hw)
  float* bufB  = bufA + (size_t)N * DH;         // N x 128 (agg / h)
  float* fused = bufB + (size_t)N * DH;         // N x 192

  // --- degrees: deg = 1 + sum over incoming edges; dis = rsqrt(deg)
  fill_kernel<<<cdiv(N, 256), 256, 0, stream>>>(dis, N, 1.0f);
  deg_accum_kernel<<<cdiv(E, 256), 256, 0, stream>>>(dstI, dis, E);
  rsqrt_kernel<<<cdiv(N, 256), 256, 0, stream>>>(dis, N);

  const int gemmGrid = cdiv(N, 128);
  const int nodeWave = cdiv((long long)N * 32, 256);  // wave-per-node
  const int node128  = cdiv((long long)N * (DH / 4), 256);
  const int node64   = cdiv((long long)N * (DOUT / 4), 256);
  const int edge128  = cdiv((long long)E * (DH / 4), 256);
  const int edge64   = cdiv((long long)E * (DOUT / 4), 256);

  for (int s = 0; s < S; ++s) {
    // layer 0: hw = x @ W0[s]; agg = hw*d^2 + b0; agg += edges; relu+LN
    gemm_wmma_kernel<DIN, DH><<<gemmGrid, 256, 0, stream>>>(
        x, W0 + (size_t)s * DIN * DH, nullptr, bufA, N);
    init_selfloop_kernel<DH><<<node128, 256, 0, stream>>>(
        bufB, bufA, dis, b0 + s * DH, N);
    edge_scatter_kernel<DH><<<edge128, 256, 0, stream>>>(
        (const float4*)bufA, srcI, dstI, dis, bufB, E);
    relu_ln_kernel<<<nodeWave, 256, 0, stream>>>(
        bufB, gamma + (s * L + 0) * DH, beta + (s * L + 0) * DH, N);

    // layer 1
    gemm_wmma_kernel<DH, DH><<<gemmGrid, 256, 0, stream>>>(
        bufB, W1 + (size_t)s * DH * DH, nullptr, bufA, N);
    init_selfloop_kernel<DH><<<node128, 256, 0, stream>>>(
        bufB, bufA, dis, b1 + s * DH, N);
    edge_scatter_kernel<DH><<<edge128, 256, 0, stream>>>(
        (const float4*)bufA, srcI, dstI, dis, bufB, E);
    relu_ln_kernel<<<nodeWave, 256, 0, stream>>>(
        bufB, gamma + (s * L + 1) * DH, beta + (s * L + 1) * DH, N);

    // layer 2 (no relu/LN): compact N x 64 result, then place into fused cols
    gemm_wmma_kernel<DH, DOUT><<<gemmGrid, 256, 0, stream>>>(
        bufB, W2 + (size_t)s * DH * DOUT, nullptr, bufA, N);
    init_selfloop_kernel<DOUT><<<node64, 256, 0, stream>>>(
        bufB, bufA, dis, b2 + s * DOUT, N);
    edge_scatter_kernel<DOUT><<<edge64, 256, 0, stream>>>(
        (const float4*)bufA, srcI, dstI, dis, bufB, E);
    copy_to_fused_kernel<<<cdiv((long long)N * 16, 256), 256, 0, stream>>>(
        bufB, fused, N, s * DOUT);
  }

  // fusion: out = fused @ Wf + bf
  gemm_wmma_kernel<S * DOUT, DOUT><<<gemmGrid, 256, 0, stream>>>(
      fused, Wf, bf, out, N);
}